// VectorQuantEMA_31499290148984
// MI455X (gfx1250) — compile-verified
//
#include <hip/hip_runtime.h>

// ---------------------------------------------------------------------------
// VQ-VAE EMA vector quantization for MI455X (gfx1250, wave32, WMMA)
//   z:        [8,16,128,256] f32  -> N=16384 rows, D=256
//   codebook: [8192,256]     f32  -> K=8192
// Outputs (f32, concatenated): z_q | loss | idx(as f32) | new_codebook |
//                              new_count | new_mean
// Score GEMM on V_WMMA_F32_16X16X4_F32; B operand double-buffered into LDS
// with GLOBAL_LOAD_ASYNC_TO_LDS_B128 (ASYNCcnt-tracked, no VGPR landing).
// ---------------------------------------------------------------------------

typedef float v2f __attribute__((ext_vector_type(2)));
typedef float v8f __attribute__((ext_vector_type(8)));

#define KCODES     8192
#define DIM        256
#define NROWS      16384
#define DECAY      0.99f
#define ONE_MINUS  0.01f
#define EPS        1e-5f

// ---- d_out layout (float offsets) ----
#define ZQ_OFF     0
#define LOSS_OFF   (ZQ_OFF + NROWS * DIM)          // 4194304
#define IDX_OFF    (LOSS_OFF + 1)                  // 4194305
#define NCB_OFF    (IDX_OFF + NROWS)               // 4210689
#define NCNT_OFF   (NCB_OFF + KCODES * DIM)        // 6307841
#define NMEAN_OFF  (NCNT_OFF + KCODES)             // 6316033

// ---- workspace layout (float offsets) ----
#define WS_CNORM   0                               // K floats
#define WS_COUNTS  (WS_CNORM + KCODES)             // K floats
#define WS_SCAL    (WS_COUNTS + KCODES)            // [0]=n_sum, [1]=sse
#define WS_MEAN    (WS_SCAL + 16)                  // K*D floats

// ---------------------------------------------------------------------------
// async copy helpers (inline asm: portable across ROCm7.2 / amdgpu-toolchain)
// ---------------------------------------------------------------------------
#define ASYNC_CP(OFF)                                                        \
  asm volatile("global_load_async_to_lds_b128 %0, %1, off offset:" #OFF      \
               :: "v"(ldsa), "v"(ga) : "memory")

// flat 16 KB tile copy: 32 lanes x 32 chunks x 16 B, stride 512 B
__device__ __forceinline__ void stage_tile(uint32_t ldsa, const float* ga) {
  ASYNC_CP(0);     ASYNC_CP(512);   ASYNC_CP(1024);  ASYNC_CP(1536);
  ASYNC_CP(2048);  ASYNC_CP(2560);  ASYNC_CP(3072);  ASYNC_CP(3584);
  ASYNC_CP(4096);  ASYNC_CP(4608);  ASYNC_CP(5120);  ASYNC_CP(5632);
  ASYNC_CP(6144);  ASYNC_CP(6656);  ASYNC_CP(7168);  ASYNC_CP(7680);
  ASYNC_CP(8192);  ASYNC_CP(8704);  ASYNC_CP(9216);  ASYNC_CP(9728);
  ASYNC_CP(10240); ASYNC_CP(10752); ASYNC_CP(11264); ASYNC_CP(11776);
  ASYNC_CP(12288); ASYNC_CP(12800); ASYNC_CP(13312); ASYNC_CP(13824);
  ASYNC_CP(14336); ASYNC_CP(14848); ASYNC_CP(15360); ASYNC_CP(15872);
}

__device__ __forceinline__ void wait_async_le32() {
  asm volatile("s_wait_asynccnt 0x20" ::: "memory");
}
__device__ __forceinline__ void wait_async_0() {
  asm volatile("s_wait_asynccnt 0x0" ::: "memory");
}

// ---------------------------------------------------------------------------
// 0) zero the atomic accumulators (must run every call: graph replay)
// ---------------------------------------------------------------------------
__global__ __launch_bounds__(256) void vq_init_kernel(float* __restrict__ ws) {
  const int stride = gridDim.x * blockDim.x;
  int i = blockIdx.x * blockDim.x + threadIdx.x;
  for (int j = i; j < KCODES * DIM; j += stride) ws[WS_MEAN + j] = 0.0f;
  for (int j = i; j < KCODES; j += stride)       ws[WS_COUNTS + j] = 0.0f;
  if (i < 16) ws[WS_SCAL + i] = 0.0f;
}

// ---------------------------------------------------------------------------
// 1) per-code squared norms  ||c||^2   (one block of 256 per code row)
// ---------------------------------------------------------------------------
__global__ __launch_bounds__(256) void vq_cnorm_kernel(
    const float* __restrict__ cb, float* __restrict__ cnorm) {
  const int k   = blockIdx.x;
  const int tid = threadIdx.x;
  float v = cb[k * DIM + tid];
  float s = v * v;
  #pragma unroll
  for (int off = 16; off >= 1; off >>= 1) s += __shfl_xor(s, off, 32);
  __shared__ float sp[8];
  if ((tid & 31) == 0) sp[tid >> 5] = s;
  __syncthreads();
  if (tid == 0) {
    float t = 0.0f;
    #pragma unroll
    for (int w = 0; w < 8; ++w) t += sp[w];
    cnorm[k] = t;
  }
}

// ---------------------------------------------------------------------------
// 2) main kernel: 16-row tile per block (8 waves split K), fp32 WMMA scores,
//    async-to-LDS double-buffered codebook tiles, argmin, z_q, SSE, scatter
// ---------------------------------------------------------------------------
__global__ __launch_bounds__(256) void vq_argmin_kernel(
    const float* __restrict__ z,
    const float* __restrict__ cb,
    const float* __restrict__ cnorm,
    float* __restrict__ out_zq,
    float* __restrict__ out_idx_f,
    float* __restrict__ counts,
    float* __restrict__ mean_acc,
    float* __restrict__ scal) {

  const int tid  = threadIdx.x;
  const int lane = tid & 31;
  const int wave = tid >> 5;
  const int lrow = lane & 15;   // row (A) / col (B,C) within 16-tile
  const int hi   = lane >> 4;   // half-wave select
  const int rowbase = blockIdx.x * 16;

  // per-wave double buffers for codebook tiles: 8 waves x 2 x 4096 floats
  __shared__ float sbuf[8 * 2 * 4096];              // 256 KB
  const uint32_t wbase =
      (uint32_t)(uintptr_t)(&sbuf[0]) + (uint32_t)wave * 32768u +
      (uint32_t)lane * 16u;

  // --- A fragments: 16x256 tile, register-resident (64 x v2f per lane) ---
  // f32 16x4 A layout: lane<16 -> k={0,1}, lane>=16 -> k={2,3} of each chunk.
  const float* zrow = z + (size_t)(rowbase + lrow) * DIM;
  v2f afrag[64];
  #pragma unroll
  for (int kk = 0; kk < 64; ++kk)
    afrag[kk] = *(const v2f*)(zrow + 4 * kk + 2 * hi);

  // running per-lane argmin of (||c||^2 - 2 x.c); lane holds 8 rows (r+8*hi)
  float bestv[8];
  int   besti[8];
  #pragma unroll
  for (int r = 0; r < 8; ++r) { bestv[r] = 3.4e38f; besti[r] = 0x7fffffff; }

  // prologue: stage first tile (jt = wave) into buffer 0
  stage_tile(wbase, cb + (size_t)(wave * 16) * DIM + lane * 4);

  for (int t = 0; t < 64; ++t) {
    const int jt    = wave + 8 * t;          // this wave's K-tile
    const int cbase = jt * 16;

    if (t + 1 < 64) {
      // stage next tile into the other buffer, then wait until the
      // previous batch (32 async ops) has fully landed.
      const int jn = jt + 8;
      stage_tile(wbase + (uint32_t)((t + 1) & 1) * 16384u,
                 cb + (size_t)(jn * 16) * DIM + lane * 4);
      wait_async_le32();
    } else {
      wait_async_0();
    }

    const float* bufp = &sbuf[wave * 8192 + (t & 1) * 4096];
    const float* brow = bufp + lrow * DIM + 2 * hi;   // lane's B column base

    v8f acc = {0.f, 0.f, 0.f, 0.f, 0.f, 0.f, 0.f, 0.f};
    #pragma unroll
    for (int kk = 0; kk < 64; ++kk) {
      v2f b = *(const v2f*)(brow + 4 * kk);           // ds_load_b64
      acc = __builtin_amdgcn_wmma_f32_16x16x4_f32(
          /*neg_a=*/false, afrag[kk], /*neg_b=*/false, b,
          /*c_mod=*/(short)0, acc, /*reuse_a=*/false, /*reuse_b=*/false);
    }

    const float cn  = cnorm[cbase + lrow];
    const int  code = cbase + lrow;
    #pragma unroll
    for (int r = 0; r < 8; ++r) {
      float d = cn - 2.0f * acc[r];
      if (d < bestv[r] || (d == bestv[r] && code < besti[r])) {
        bestv[r] = d; besti[r] = code;
      }
    }
  }

  // --- xor-butterfly min-with-index across the 16 lanes of each half ---
  #pragma unroll
  for (int off = 8; off >= 1; off >>= 1) {
    #pragma unroll
    for (int r = 0; r < 8; ++r) {
      float ov = __shfl_xor(bestv[r], off, 32);
      int   oi = __shfl_xor(besti[r], off, 32);
      if (ov < bestv[r] || (ov == bestv[r] && oi < besti[r])) {
        bestv[r] = ov; besti[r] = oi;
      }
    }
  }

  // --- cross-wave reduce via LDS ---
  __shared__ float s_val[8][16];
  __shared__ int   s_idx[8][16];
  __shared__ int   s_fin[16];
  if (lane == 0 || lane == 16) {
    #pragma unroll
    for (int r = 0; r < 8; ++r) {
      s_val[wave][r + 8 * hi] = bestv[r];
      s_idx[wave][r + 8 * hi] = besti[r];
    }
  }
  __syncthreads();

  if (tid < 16) {
    float bv = s_val[0][tid];
    int   bi = s_idx[0][tid];
    #pragma unroll
    for (int w = 1; w < 8; ++w) {
      float ov = s_val[w][tid];
      int   oi = s_idx[w][tid];
      if (ov < bv || (ov == bv && oi < bi)) { bv = ov; bi = oi; }
    }
    s_fin[tid] = bi;
    out_idx_f[rowbase + tid] = (float)bi;          // idx output (as f32)
    atomicAdd(&counts[bi], 1.0f);                  // segment count
  }
  __syncthreads();

  // --- z_q write, SSE partial, EMA mean scatter (tid == dim index) ---
  float ssum = 0.0f;
  for (int r = 0; r < 16; ++r) {
    const int fi = s_fin[r];
    const float zq = cb[(size_t)fi * DIM + tid];
    const float zv = z[(size_t)(rowbase + r) * DIM + tid];
    out_zq[(size_t)(rowbase + r) * DIM + tid] = zq; // z_q_st == z_q in value
    const float d = zq - zv;
    ssum += d * d;
    atomicAdd(&mean_acc[(size_t)fi * DIM + tid], zv);
  }
  #pragma unroll
  for (int off = 16; off >= 1; off >>= 1) ssum += __shfl_xor(ssum, off, 32);
  if (lane == 0) atomicAdd(&scal[1], ssum);
}

// ---------------------------------------------------------------------------
// 3) new_count = decay*ema_count + (1-decay)*counts ;  n = sum(new_count)
// ---------------------------------------------------------------------------
__global__ __launch_bounds__(256) void vq_newcount_kernel(
    const float* __restrict__ ema_count, const float* __restrict__ counts,
    float* __restrict__ out_newcount, float* __restrict__ scal) {
  const int i = blockIdx.x * 256 + threadIdx.x;
  float nc = DECAY * ema_count[i] + ONE_MINUS * counts[i];
  out_newcount[i] = nc;
  float s = nc;
  #pragma unroll
  for (int off = 16; off >= 1; off >>= 1) s += __shfl_xor(s, off, 32);
  if ((threadIdx.x & 31) == 0) atomicAdd(&scal[0], s);
}

// ---------------------------------------------------------------------------
// 4) new_mean, new_codebook, loss
// ---------------------------------------------------------------------------
__global__ __launch_bounds__(256) void vq_finalize_kernel(
    const float* __restrict__ ema_mean, const float* __restrict__ mean_acc,
    const float* __restrict__ out_newcount, const float* __restrict__ scal,
    float* __restrict__ out_newmean, float* __restrict__ out_newcb,
    float* __restrict__ out_loss) {
  const int k = blockIdx.x;
  const int d = threadIdx.x;
  const size_t o = (size_t)k * DIM + d;
  const float nm = DECAY * ema_mean[o] + ONE_MINUS * mean_acc[o];
  out_newmean[o] = nm;
  const float n  = scal[0];
  const float nc = out_newcount[k];
  const float cluster = (nc + EPS) / (n + (float)KCODES * EPS) * n;
  out_newcb[o] = nm / cluster;
  if (k == 0 && d == 0)
    out_loss[0] = 1.25f * scal[1] / (float)(NROWS * DIM);
}

// ---------------------------------------------------------------------------
extern "C" void kernel_launch(void* const* d_in, const int* in_sizes, int n_in,
                              void* d_out, int out_size, void* d_ws, size_t ws_size,
                              hipStream_t stream) {
  const float* z         = (const float*)d_in[0];
  const float* codebook  = (const float*)d_in[1];
  const float* ema_count = (const float*)d_in[2];
  const float* ema_mean  = (const float*)d_in[3];

  float* out = (float*)d_out;
  float* ws  = (float*)d_ws;

  float* cnorm    = ws + WS_CNORM;
  float* counts   = ws + WS_COUNTS;
  float* scal     = ws + WS_SCAL;
  float* mean_acc = ws + WS_MEAN;

  vq_init_kernel<<<2048, 256, 0, stream>>>(ws);
  vq_cnorm_kernel<<<KCODES, 256, 0, stream>>>(codebook, cnorm);
  vq_argmin_kernel<<<NROWS / 16, 256, 0, stream>>>(
      z, codebook, cnorm,
      out + ZQ_OFF, out + IDX_OFF, counts, mean_acc, scal);
  vq_newcount_kernel<<<KCODES / 256, 256, 0, stream>>>(
      ema_count, counts, out + NCNT_OFF, scal);
  vq_finalize_kernel<<<KCODES, 256, 0, stream>>>(
      ema_mean, mean_acc, out + NCNT_OFF, scal,
      out + NMEAN_OFF, out + NCB_OFF, out + LOSS_OFF);
}